// LogEntmaxBisect_62551903699248
// MI455X (gfx1250) — compile-verified
//
#include <hip/hip_runtime.h>
#include <stdint.h>

// log(entmax_bisect(X)) row-wise, alpha = 1.5, d = 32000, rows = 4096.
// One block (1024 threads = 32 waves) per row.
//  - Row staged HBM -> LDS once via CDNA5 async copies (global_load_async_to_lds_b128).
//  - Each thread keeps 32 strided fp32 values in VGPRs; all 50 bisection sweeps
//    run from registers (single HBM pass total: 512 MB in + 512 MB out).
//  - Bisection early-exits (uniformly) once tau_lo + dm == tau_lo in fp32, which is
//    bit-equivalent to running all 50 reference iterations.

#define D_COLS 32000
#define BLOCK 1024
#define EPT 32                    // ceil(32000 / 1024)
#define B128_PER_ROW (D_COLS / 4) // 8000 16-byte chunks per row

__device__ __forceinline__ float wave_reduce_sum(float v) {
#pragma unroll
  for (int m = 16; m > 0; m >>= 1) v += __shfl_xor(v, m, 32);
  return v;
}

__device__ __forceinline__ float wave_reduce_max(float v) {
#pragma unroll
  for (int m = 16; m > 0; m >>= 1) v = fmaxf(v, __shfl_xor(v, m, 32));
  return v;
}

// Full-block reduce with broadcast of the result to every thread.
__device__ __forceinline__ float block_reduce(float v, float* scratch, int tid, bool is_max) {
  v = is_max ? wave_reduce_max(v) : wave_reduce_sum(v);
  const int wave = tid >> 5;
  const int lane = tid & 31;
  if (lane == 0) scratch[wave] = v;
  __syncthreads();
  if (wave == 0) {
    float w = scratch[lane];           // 32 waves -> 32 partials
    w = is_max ? wave_reduce_max(w) : wave_reduce_sum(w);
    if (lane == 0) scratch[32] = w;
  }
  __syncthreads();
  const float r = scratch[32];
  __syncthreads();                     // scratch is reused by the next reduction
  return r;
}

__global__ void __launch_bounds__(BLOCK)
log_entmax_kernel(const float* __restrict__ X, float* __restrict__ out, int rows) {
  extern __shared__ float smem[];
  float* scratch = smem;        // 64 floats reserved (33 used)
  float* rowbuf  = smem + 64;   // 32000 floats, 256B-aligned

  const int tid = threadIdx.x;
  const int row = blockIdx.x;
  if (row >= rows) return;      // uniform per block

  // ---- Stage the row into LDS with CDNA5 async global->LDS B128 copies ----
  const uint64_t gbase    = (uint64_t)(uintptr_t)(X + (size_t)row * D_COLS);
  const uint32_t lds_base = (uint32_t)(uintptr_t)rowbuf;   // low 32 bits of the LDS aperture
#pragma unroll
  for (int j = 0; j < 8; ++j) {
    const int chunk = tid + j * BLOCK;                     // b128 index within row
    if (chunk < B128_PER_ROW) {
      const uint32_t byte = (uint32_t)chunk * 16u;
      const uint32_t ldsa = lds_base + byte;
      asm volatile("global_load_async_to_lds_b128 %0, %1, %2"
                   :: "v"(ldsa), "v"(byte), "s"(gbase)
                   : "memory");
    }
  }
  asm volatile("s_wait_asynccnt 0x0" ::: "memory");
  __syncthreads();

  // ---- Pull 32 strided values per thread into registers (bank-conflict-free) ----
  // Xs = (alpha-1) * X = 0.5 * X. Pad out-of-range slots with -inf (relu kills them).
  float xs[EPT];
#pragma unroll
  for (int k = 0; k < EPT; ++k) {
    const int i = tid + k * BLOCK;
    xs[k] = (i < D_COLS) ? 0.5f * rowbuf[i] : -__builtin_huge_valf();
  }

  // ---- Row max ----
  float m = -__builtin_huge_valf();
#pragma unroll
  for (int k = 0; k < EPT; ++k) m = fmaxf(m, xs[k]);
  m = block_reduce(m, scratch, tid, true);

  float tau_lo = m - 1.0f;                             // gp_inv(1, 1.5)
  const float tau_hi = m - 0.005590169943749474f;      // gp_inv(1/d, 1.5) = sqrt(1/32000)

  // f_lo = sum(relu(Xs - tau_lo)^2) - 1  (fixed through the whole bisection)
  float acc = 0.0f;
#pragma unroll
  for (int k = 0; k < EPT; ++k) {
    float t = fmaxf(xs[k] - tau_lo, 0.0f);
    acc = fmaf(t, t, acc);
  }
  const float f_lo = block_reduce(acc, scratch, tid, false) - 1.0f;

  // ---- Bisection (50 iterations; uniform break once fp32 fixpoint reached) ----
  float dm = tau_hi - tau_lo;
  float tau_m = tau_lo;
#pragma unroll 1
  for (int it = 0; it < 50; ++it) {
    dm *= 0.5f;
    tau_m = tau_lo + dm;
    if (tau_m == tau_lo) break;   // remaining reference iterations are exact no-ops
    float a = 0.0f;
#pragma unroll
    for (int k = 0; k < EPT; ++k) {
      float t = fmaxf(xs[k] - tau_m, 0.0f);
      a = fmaf(t, t, a);
    }
    const float f_m = block_reduce(a, scratch, tid, false) - 1.0f;
    if (f_m * f_lo >= 0.0f) tau_lo = tau_m;
  }

  // ---- p = relu(Xs - tau_m)^2 ; normalize; redundant renorm; log ----
  float S = 0.0f;
#pragma unroll
  for (int k = 0; k < EPT; ++k) {
    float t = fmaxf(xs[k] - tau_m, 0.0f);
    S = fmaf(t, t, S);
  }
  S = block_reduce(S, scratch, tid, false);
  const float invS = 1.0f / S;

  float T = 0.0f;
#pragma unroll
  for (int k = 0; k < EPT; ++k) {
    float t = fmaxf(xs[k] - tau_m, 0.0f);
    T += (t * t) * invS;
  }
  T = block_reduce(T, scratch, tid, false);
  const float invST = invS * (1.0f / T);

  float* orow = out + (size_t)row * D_COLS;
#pragma unroll
  for (int k = 0; k < EPT; ++k) {
    const int i = tid + k * BLOCK;
    if (i < D_COLS) {
      float t = fmaxf(xs[k] - tau_m, 0.0f);
      orow[i] = logf((t * t) * invST);   // log(0) -> -inf, matching jnp.log
    }
  }
}

extern "C" void kernel_launch(void* const* d_in, const int* in_sizes, int n_in,
                              void* d_out, int out_size, void* d_ws, size_t ws_size,
                              hipStream_t stream) {
  const float* X = (const float*)d_in[0];
  float* out = (float*)d_out;
  const int rows = in_sizes[0] / D_COLS;
  const size_t shmem = (size_t)(64 + D_COLS) * sizeof(float);  // 128,256 bytes
  (void)hipFuncSetAttribute((const void*)log_entmax_kernel,
                            hipFuncAttributeMaxDynamicSharedMemorySize, (int)shmem);
  log_entmax_kernel<<<rows, BLOCK, shmem, stream>>>(X, out, rows);
}